// RailCostBenefitLoss_35270271434971
// MI455X (gfx1250) — compile-verified
//
#include <hip/hip_runtime.h>
#include <stdint.h>

// ---------------------------------------------------------------------------
// RailCostBenefitLoss for MI455X (gfx1250, wave32).
//
//   fw_init        : w0 = dist/(soft+DELTA), zero diagonal            (1 launch)
//   fw_step_fused  : FUSE=8 soft-Floyd-Warshall pivots per launch     (64 launches)
//   finalize       : fused loss terms -> per-block partial sums       (1 launch)
//   final_sum      : deterministic tree-sum of partials -> d_out[0]   (1 launch)
//
// Pivot rows are staged into LDS via gfx1250 global_load_async_to_lds_b128
// (ASYNCcnt) and evolved in-block so 8 sequential pivots need 1 dispatch.
// ---------------------------------------------------------------------------

#define NMAT 512
#define FUSE 8

__device__ __forceinline__ float fast_exp2(float x) {
#if __has_builtin(__builtin_amdgcn_exp2f)
  return __builtin_amdgcn_exp2f(x);        // v_exp_f32
#else
  return exp2f(x);
#endif
}

__device__ __forceinline__ float fast_log2(float x) {
#if __has_builtin(__builtin_amdgcn_logf)
  return __builtin_amdgcn_logf(x);         // v_log_f32 (log base 2)
#else
  return log2f(x);
#endif
}

// softmin_gamma(a,b) = -g*logaddexp(-a/g, -b/g)
//                    = min(a,b) - g*ln2 * log2(1 + 2^(-|a-b| / (g*ln2)))
// GAMMA = 0.1
__device__ __forceinline__ float softmin_g(float a, float b) {
  const float C1 = 14.426950408889634f;   // 1 / (GAMMA * ln2)
  const float C2 = 0.06931471805599453f;  // GAMMA * ln2
  float m = fminf(a, b);
  float d = fabsf(a - b);
  float z = fast_exp2(-d * C1);
  return m - C2 * fast_log2(1.0f + z);
}

// ---------------------------------------------------------------------------
// Init: W = distances / (soft_adj + DELTA); diagonal zeroed.
__global__ __launch_bounds__(256) void fw_init(const float* __restrict__ soft,
                                               const float* __restrict__ dist,
                                               float* __restrict__ w0) {
  const float DELTA = 1e-4f;
  int e = (blockIdx.x * 256 + threadIdx.x) * 4;
  int i = e >> 9;
  int j = e & (NMAT - 1);
  float4 s = *(const float4*)(soft + e);
  float4 d = *(const float4*)(dist + e);
  float4 w;
  w.x = d.x / (s.x + DELTA);
  w.y = d.y / (s.y + DELTA);
  w.z = d.z / (s.z + DELTA);
  w.w = d.w / (s.w + DELTA);
  if (i == j + 0) w.x = 0.0f;
  if (i == j + 1) w.y = 0.0f;
  if (i == j + 2) w.z = 0.0f;
  if (i == j + 3) w.w = 0.0f;
  *(float4*)(w0 + e) = w;
}

// ---------------------------------------------------------------------------
// FUSE consecutive FW pivots (k .. k+FUSE-1) in one dispatch.
// Block b owns row i = b; 128 threads; thread t owns columns 4t..4t+3 of
// every staged row. Pivot rows live in LDS and are evolved in-block.
__global__ __launch_bounds__(128) void fw_step_fused(const float* __restrict__ src,
                                                     float* __restrict__ dst,
                                                     int k) {
  __shared__ float R[FUSE][NMAT];          // staged pivot rows, 16 KB
  const int i  = blockIdx.x;
  const int t  = threadIdx.x;
  const int j0 = t * 4;

  // Async stage: FUSE rows x (128 lanes x 16B) into LDS (ASYNCcnt).
#pragma unroll
  for (int m = 0; m < FUSE; ++m) {
    uint32_t lds_addr = (uint32_t)(uintptr_t)(&R[m][j0]);
    uint64_t gaddr    = (uint64_t)(uintptr_t)(src + (size_t)(k + m) * NMAT + j0);
    asm volatile("global_load_async_to_lds_b128 %0, %1, off"
                 :: "v"(lds_addr), "v"(gaddr)
                 : "memory");
  }

  // Overlap with the async DMA: own row chunk + pivot-column scalars.
  float4 av = *(const float4*)(src + (size_t)i * NMAT + j0);
  float a[4] = {av.x, av.y, av.z, av.w};
  float c[FUSE];                           // c[m] = w_cur[i][k+m] (uniform)
#pragma unroll
  for (int m = 0; m < FUSE; ++m) c[m] = src[(size_t)i * NMAT + (k + m)];

  asm volatile("s_wait_asynccnt 0" ::: "memory");
  __syncthreads();                         // rows visible to all waves

#pragma unroll
  for (int m = 0; m < FUSE; ++m) {
    // ---- read phase: snapshot pre-update state ----
    float pm[4];                           // pivot row R[m] at own columns
#pragma unroll
    for (int q = 0; q < 4; ++q) pm[q] = R[m][j0 + q];
    float sR[FUSE];                        // R[r][k+m], r>m (uniform reads)
#pragma unroll
    for (int r = m + 1; r < FUSE; ++r) sR[r] = R[r][k + m];
    float sP[FUSE];                        // R[m][k+m'], m'>m (uniform reads)
#pragma unroll
    for (int m2 = m + 1; m2 < FUSE; ++m2) sP[m2] = R[m][k + m2];
    __syncthreads();                       // reads done before in-place writes

    // ---- write phase ----
    // own row: w'[i][j] = softmin(w[i][j], w[i][p] + w[p][j])
#pragma unroll
    for (int q = 0; q < 4; ++q) a[q] = softmin_g(a[q], c[m] + pm[q]);
    // future pivot rows: w'[k+r][j] = softmin(w[k+r][j], w[k+r][p] + w[p][j])
#pragma unroll
    for (int r = m + 1; r < FUSE; ++r) {
      float s = sR[r];
#pragma unroll
      for (int q = 0; q < 4; ++q)
        R[r][j0 + q] = softmin_g(R[r][j0 + q], s + pm[q]);
    }
    // pivot-column scalars: w'[i][k+m'] = softmin(w[i][k+m'], w[i][p] + w[p][k+m'])
#pragma unroll
    for (int m2 = m + 1; m2 < FUSE; ++m2) c[m2] = softmin_g(c[m2], c[m] + sP[m2]);
    __syncthreads();                       // writes visible before next step
  }

  float4 out = {a[0], a[1], a[2], a[3]};
  *(float4*)(dst + (size_t)i * NMAT + j0) = out;
}

// ---------------------------------------------------------------------------
// Fused final pass: all four loss terms, per-block partial sums.
__global__ __launch_bounds__(256) void finalize(const float* __restrict__ soft,
                                                const float* __restrict__ orig,
                                                const float* __restrict__ dist,
                                                const float* __restrict__ flow,
                                                const float* __restrict__ wfin,
                                                const int* __restrict__ epoch_p,
                                                float* __restrict__ partials) {
  const float UTILITY_SCALE = -0.01f;
  const float PRIORITY_RAIL = 0.5f;
  const float MASK_LEVEL    = 10000.0f;
  const float LEVELS[4]     = {0.0f, 0.05f, 0.1f, 1.0f};

  int ep = *epoch_p;  // scalar, uniform
  int li = (ep >= 0) + (ep >= 10) + (ep >= 50);   // bisect_right([0,10,50], ep)
  float escale = LEVELS[li];

  int e0 = (blockIdx.x * 256 + threadIdx.x) * 4;
  float acc = 0.0f;
#pragma unroll
  for (int tidx = 0; tidx < 4; ++tidx) {
    int e = e0 + tidx;
    int i = e >> 9;
    int j = e & (NMAT - 1);
    float s  = soft[e];
    float og = orig[e];
    float di = dist[e];
    float fl = flow[e];
    float sp = wfin[e];

    // loss_cost
    acc += s * di;

    // utility_gain (elu(x)+1)
    float ur = expf(UTILITY_SCALE * sp * PRIORITY_RAIL);
    float ub = expf(UTILITY_SCALE * di);
    float ch = ur / (ur + ub);
    float ds = di - PRIORITY_RAIL * sp;
    float ug = fl * ch * ds;
    float el = (ug > 0.0f) ? ug : expm1f(ug);
    acc += el + 1.0f;

    // entropy: (soft * (eye - soft))^2, scaled
    float eye = (i == j) ? 1.0f : 0.0f;
    float ent = s * (eye - s);
    acc += escale * ent * ent;

    // mask loss
    acc += MASK_LEVEL * s * (1.0f - og);
  }

  __shared__ float red[256];
  red[threadIdx.x] = acc;
  __syncthreads();
#pragma unroll
  for (int stp = 128; stp > 0; stp >>= 1) {
    if (threadIdx.x < stp) red[threadIdx.x] += red[threadIdx.x + stp];
    __syncthreads();
  }
  if (threadIdx.x == 0) partials[blockIdx.x] = red[0];
}

// 1 block x 256 threads: sum the 256 partials deterministically.
__global__ __launch_bounds__(256) void final_sum(const float* __restrict__ partials,
                                                 float* __restrict__ out) {
  __shared__ float red[256];
  red[threadIdx.x] = partials[threadIdx.x];
  __syncthreads();
#pragma unroll
  for (int stp = 128; stp > 0; stp >>= 1) {
    if (threadIdx.x < stp) red[threadIdx.x] += red[threadIdx.x + stp];
    __syncthreads();
  }
  if (threadIdx.x == 0) out[0] = red[0];
}

// ---------------------------------------------------------------------------
extern "C" void kernel_launch(void* const* d_in, const int* in_sizes, int n_in,
                              void* d_out, int out_size, void* d_ws, size_t ws_size,
                              hipStream_t stream) {
  (void)in_sizes; (void)n_in; (void)out_size; (void)ws_size;
  const float* soft  = (const float*)d_in[0];
  const float* orig  = (const float*)d_in[1];
  const float* dist  = (const float*)d_in[2];
  const float* flow  = (const float*)d_in[3];
  const int*   epoch = (const int*)d_in[4];

  float* wsA      = (float*)d_ws;             // 512*512 floats
  float* wsB      = wsA + NMAT * NMAT;        // 512*512 floats
  float* partials = wsB + NMAT * NMAT;        // 256 floats

  fw_init<<<256, 256, 0, stream>>>(soft, dist, wsA);

  float* src = wsA;
  float* dst = wsB;
  for (int k = 0; k < NMAT; k += FUSE) {       // 64 fused dispatches
    fw_step_fused<<<NMAT, 128, 0, stream>>>(src, dst, k);
    float* tmp = src; src = dst; dst = tmp;
  }
  // 64 swaps (even) -> result back in wsA (== src after final swap)

  finalize<<<256, 256, 0, stream>>>(soft, orig, dist, flow, src, epoch, partials);
  final_sum<<<1, 256, 0, stream>>>(partials, (float*)d_out);
}